// SensorGAT_65240553226420
// MI455X (gfx1250) — compile-verified
//
#include <hip/hip_runtime.h>
#include <hip/hip_bf16.h>
#include <math.h>

typedef __attribute__((ext_vector_type(2))) float v2f;
typedef __attribute__((ext_vector_type(8))) float v8f;

#define NNODES 50000
#define EORIG  800000
#define ETOT   (EORIG + NNODES)
#define GB     64

// ---------------------------------------------------------------------------
// fp32 WMMA GEMM: out[N x ncols] = A[N x K] @ W[K x ncols]
// One wave (32 lanes) computes one 16x16 tile, K-loop in steps of 4 using
// V_WMMA_F32_16X16X4_F32.
//   A frag (16x4):  lane l holds row r=l&15, elems j at k = k0 + 2*(l>>4) + j
//   B frag (4x16):  lane l holds col c=l&15, elems j at k = k0 + 2*(l>>4) + j
//   C/D (16x16):    elem v = D[row v + 8*(l>>4)][col l&15]
// ---------------------------------------------------------------------------
__global__ void gemm_wmma_kernel(const float* __restrict__ A,
                                 const float* __restrict__ W,
                                 float* __restrict__ out,
                                 int K, int ncols) {
  const int lane = threadIdx.x;          // 0..31 (one wave per threadIdx.y row)
  const int r    = lane & 15;
  const int kg   = lane >> 4;            // 0 or 1
  const int rowBase = blockIdx.x * 16;
  const int colTile = blockIdx.y * blockDim.y + threadIdx.y;
  const int colBase = colTile * 16;
  const float* arow = A + (size_t)(rowBase + r) * K;

  v8f acc = {};
  for (int k0 = 0; k0 < K; k0 += 4) {
    const int ka = k0 + 2 * kg;
    v2f av; av.x = arow[ka];                              av.y = arow[ka + 1];
    v2f bv; bv.x = W[(size_t)ka * ncols + colBase + r];   bv.y = W[(size_t)(ka + 1) * ncols + colBase + r];
    acc = __builtin_amdgcn_wmma_f32_16x16x4_f32(
        /*neg_a=*/false, av, /*neg_b=*/false, bv,
        /*c_mod=*/(short)0, acc, /*reuse_a=*/false, /*reuse_b=*/false);
  }
#pragma unroll
  for (int v = 0; v < 8; ++v) {
    const int m = v + 8 * kg;
    out[(size_t)(rowBase + m) * ncols + colBase + r] = acc[v];
  }
}

// ---------------------------------------------------------------------------
// es[n,h] = sum_c h[n, h*64+c] * a_s[h*64+c]   (and same for ed / a_d)
// One wave per (n,h); C is fixed at 64.
// ---------------------------------------------------------------------------
__global__ void scores_kernel(const float* __restrict__ h,
                              const float* __restrict__ a_s,
                              const float* __restrict__ a_d,
                              float* __restrict__ es, float* __restrict__ ed,
                              int nWaves, int H) {
  const int gtid = blockIdx.x * blockDim.x + threadIdx.x;
  const int wave = gtid >> 5;
  const int lane = gtid & 31;
  if (wave >= nWaves) return;
  const int n  = wave / H;
  const int hh = wave - n * H;
  const float* hp  = h + (size_t)n * H * 64 + hh * 64;
  const float* asp = a_s + hh * 64;
  const float* adp = a_d + hh * 64;
  float s0 = hp[lane] * asp[lane] + hp[lane + 32] * asp[lane + 32];
  float s1 = hp[lane] * adp[lane] + hp[lane + 32] * adp[lane + 32];
#pragma unroll
  for (int off = 16; off > 0; off >>= 1) {
    s0 += __shfl_xor(s0, off, 32);
    s1 += __shfl_xor(s1, off, 32);
  }
  if (lane == 0) { es[wave] = s0; ed[wave] = s1; }
}

__device__ __forceinline__ void edge_src_dst(const int* __restrict__ ei, int e,
                                             int& s, int& d) {
  if (e < EORIG) { s = ei[e]; d = ei[EORIG + e]; }
  else           { s = e - EORIG; d = s; }      // self loops
}

__device__ __forceinline__ float leaky(float v) {
  return v > 0.0f ? v : 0.2f * v;
}

__device__ __forceinline__ void atomicMaxF32(float* addr, float v) {
  if (v >= 0.0f) atomicMax((int*)addr, __float_as_int(v));
  else           atomicMin((unsigned int*)addr, __float_as_uint(v));
}

__global__ void edge_max_kernel(const int* __restrict__ ei,
                                const float* __restrict__ es,
                                const float* __restrict__ ed,
                                float* __restrict__ mx, int H) {
  const int idx = blockIdx.x * blockDim.x + threadIdx.x;
  if (idx >= ETOT * H) return;
  const int e = idx / H, hh = idx - e * H;
  int s, d; edge_src_dst(ei, e, s, d);
  const float v = leaky(es[s * H + hh] + ed[d * H + hh]);
  atomicMaxF32(mx + d * H + hh, v);
}

__global__ void edge_denom_kernel(const int* __restrict__ ei,
                                  const float* __restrict__ es,
                                  const float* __restrict__ ed,
                                  const float* __restrict__ mx,
                                  float* __restrict__ dn, int H) {
  const int idx = blockIdx.x * blockDim.x + threadIdx.x;
  if (idx >= ETOT * H) return;
  const int e = idx / H, hh = idx - e * H;
  int s, d; edge_src_dst(ei, e, s, d);
  const float v = leaky(es[s * H + hh] + ed[d * H + hh]);
  atomicAdd(dn + d * H + hh, expf(v - mx[d * H + hh]));
}

// grid = ETOT blocks, block = H*C threads; thread c -> channel h*C + c
__global__ void edge_agg_kernel(const int* __restrict__ ei,
                                const float* __restrict__ es,
                                const float* __restrict__ ed,
                                const float* __restrict__ mx,
                                const float* __restrict__ dn,
                                const float* __restrict__ h,
                                float* __restrict__ out, int H, int C) {
  const int e = blockIdx.x;
  const int c = threadIdx.x;
  const int hh = c / C;
  int s, d; edge_src_dst(ei, e, s, d);
  const float v = leaky(es[s * H + hh] + ed[d * H + hh]);
  const float alpha = expf(v - mx[d * H + hh]) / dn[d * H + hh];
  atomicAdd(out + (size_t)d * H * C + c, h[(size_t)s * H * C + c] * alpha);
}

// in-place: x = BN(x + bias); optional ELU
__global__ void bn_act_kernel(float* __restrict__ x, const float* __restrict__ bias,
                              const float* __restrict__ g, const float* __restrict__ be,
                              const float* __restrict__ mean, const float* __restrict__ var,
                              int total, int HC, int doElu) {
  const int i = blockIdx.x * blockDim.x + threadIdx.x;
  if (i >= total) return;
  const int c = i % HC;
  float v = x[i] + bias[c];
  v = (v - mean[c]) * rsqrtf(var[c] + 1e-5f) * g[c] + be[c];
  if (doElu) v = v > 0.0f ? v : (expf(v) - 1.0f);
  x[i] = v;
}

__global__ void fill_kernel(float* __restrict__ p, float v, int n) {
  const int i = blockIdx.x * blockDim.x + threadIdx.x;
  if (i < n) p[i] = v;
}

__global__ void pool_kernel(const float* __restrict__ h, const int* __restrict__ batch,
                            float* __restrict__ pooled, float* __restrict__ cnt) {
  const int i = blockIdx.x * blockDim.x + threadIdx.x;
  if (i >= NNODES * 64) return;
  const int n = i >> 6, c = i & 63;
  const int g = batch[n];
  atomicAdd(pooled + g * 64 + c, h[i]);
  if (c == 0) atomicAdd(cnt + g, 1.0f);
}

// one block of 64 threads, thread g handles one graph end-to-end
__global__ void head_kernel(const float* __restrict__ pooled, const float* __restrict__ cnt,
                            const float* __restrict__ lw1, const float* __restrict__ lb1,
                            const float* __restrict__ lw2, const float* __restrict__ lb2,
                            float* __restrict__ out) {
  const int g = threadIdx.x;
  if (g >= GB) return;
  float pm[64];
  const float inv = 1.0f / fmaxf(cnt[g], 1.0f);
#pragma unroll 8
  for (int c = 0; c < 64; ++c) pm[c] = pooled[g * 64 + c] * inv;
  float z1[32];
  for (int j = 0; j < 32; ++j) {
    float acc = lb1[j];
    for (int c = 0; c < 64; ++c) acc += pm[c] * lw1[c * 32 + j];
    z1[j] = fmaxf(acc, 0.0f);
  }
  float z2[3];
  for (int k = 0; k < 3; ++k) {
    float acc = lb2[k];
    for (int j = 0; j < 32; ++j) acc += z1[j] * lw2[j * 3 + k];
    z2[k] = acc;
  }
  const float m = fmaxf(z2[0], fmaxf(z2[1], z2[2]));
  const float ls = m + logf(expf(z2[0] - m) + expf(z2[1] - m) + expf(z2[2] - m));
  for (int k = 0; k < 3; ++k) out[g * 3 + k] = z2[k] - ls;
}

extern "C" void kernel_launch(void* const* d_in, const int* in_sizes, int n_in,
                              void* d_out, int out_size, void* d_ws, size_t ws_size,
                              hipStream_t stream) {
  (void)in_sizes; (void)n_in; (void)out_size; (void)ws_size;
  const float* x    = (const float*)d_in[0];
  const int*   ei   = (const int*)d_in[1];
  const int*   batch= (const int*)d_in[2];
  const float* W1   = (const float*)d_in[3];
  const float* aS1  = (const float*)d_in[4];
  const float* aD1  = (const float*)d_in[5];
  const float* b1   = (const float*)d_in[6];
  const float* g1   = (const float*)d_in[7];
  const float* be1  = (const float*)d_in[8];
  const float* m1   = (const float*)d_in[9];
  const float* v1   = (const float*)d_in[10];
  const float* W2   = (const float*)d_in[11];
  const float* aS2  = (const float*)d_in[12];
  const float* aD2  = (const float*)d_in[13];
  const float* b2   = (const float*)d_in[14];
  const float* g2   = (const float*)d_in[15];
  const float* be2  = (const float*)d_in[16];
  const float* m2   = (const float*)d_in[17];
  const float* v2   = (const float*)d_in[18];
  const float* lw1  = (const float*)d_in[19];
  const float* lb1  = (const float*)d_in[20];
  const float* lw2  = (const float*)d_in[21];
  const float* lb2  = (const float*)d_in[22];
  float* out = (float*)d_out;

  // workspace layout (floats)
  float* ws   = (float*)d_ws;
  float* h1   = ws;                       // 12,800,000  (N*256)
  float* out1 = h1 + 12800000;            // 12,800,000
  float* es1  = out1 + 12800000;          // 200,000 each below
  float* ed1  = es1 + 200000;
  float* mx1  = ed1 + 200000;
  float* dn1  = mx1 + 200000;
  float* es2  = dn1 + 200000;             // 50,000 each below
  float* ed2  = es2 + 50000;
  float* mx2  = ed2 + 50000;
  float* dn2  = mx2 + 50000;
  float* pooled = dn2 + 50000;            // 4096
  float* cnt  = pooled + 4096;            // 64
  float* h2   = h1;                       // reuse (3,200,000)
  float* out2 = h1 + 3200000;             // reuse (3,200,000)

  const int B = 256;
  auto blocks = [](long long n, int b) { return (int)((n + b - 1) / b); };

  // ---- Layer 1: h1 = x @ W1 (WMMA fp32), scores, softmax-aggregate ----
  gemm_wmma_kernel<<<dim3(NNODES / 16, 4), dim3(32, 4), 0, stream>>>(x, W1, h1, 64, 256);
  scores_kernel<<<blocks((long long)NNODES * 4 * 32, B), B, 0, stream>>>(h1, aS1, aD1, es1, ed1, NNODES * 4, 4);
  fill_kernel<<<blocks(200000, B), B, 0, stream>>>(mx1, -1e30f, 200000);
  fill_kernel<<<blocks(200000, B), B, 0, stream>>>(dn1, 0.0f, 200000);
  fill_kernel<<<blocks(12800000, B), B, 0, stream>>>(out1, 0.0f, 12800000);
  edge_max_kernel<<<blocks((long long)ETOT * 4, B), B, 0, stream>>>(ei, es1, ed1, mx1, 4);
  edge_denom_kernel<<<blocks((long long)ETOT * 4, B), B, 0, stream>>>(ei, es1, ed1, mx1, dn1, 4);
  edge_agg_kernel<<<ETOT, 256, 0, stream>>>(ei, es1, ed1, mx1, dn1, h1, out1, 4, 64);
  bn_act_kernel<<<blocks(12800000, B), B, 0, stream>>>(out1, b1, g1, be1, m1, v1, 12800000, 256, 1);

  // ---- Layer 2: h2 = out1 @ W2 (WMMA fp32), scores, softmax-aggregate ----
  gemm_wmma_kernel<<<dim3(NNODES / 16, 1), dim3(32, 4), 0, stream>>>(out1, W2, h2, 256, 64);
  scores_kernel<<<blocks((long long)NNODES * 32, B), B, 0, stream>>>(h2, aS2, aD2, es2, ed2, NNODES, 1);
  fill_kernel<<<blocks(50000, B), B, 0, stream>>>(mx2, -1e30f, 50000);
  fill_kernel<<<blocks(50000, B), B, 0, stream>>>(dn2, 0.0f, 50000);
  fill_kernel<<<blocks(3200000, B), B, 0, stream>>>(out2, 0.0f, 3200000);
  edge_max_kernel<<<blocks(ETOT, B), B, 0, stream>>>(ei, es2, ed2, mx2, 1);
  edge_denom_kernel<<<blocks(ETOT, B), B, 0, stream>>>(ei, es2, ed2, mx2, dn2, 1);
  edge_agg_kernel<<<ETOT, 64, 0, stream>>>(ei, es2, ed2, mx2, dn2, h2, out2, 1, 64);
  bn_act_kernel<<<blocks(3200000, B), B, 0, stream>>>(out2, b2, g2, be2, m2, v2, 3200000, 64, 0);

  // ---- Pool + MLP head + log_softmax ----
  fill_kernel<<<blocks(4096 + 64, B), B, 0, stream>>>(pooled, 0.0f, 4096 + 64);
  pool_kernel<<<blocks((long long)NNODES * 64, B), B, 0, stream>>>(out2, batch, pooled, cnt);
  head_kernel<<<1, 64, 0, stream>>>(pooled, cnt, lw1, lb1, lw2, lb2, out);
}